// LocalStridedBlockSparsePagedAttn_13443247636953
// MI455X (gfx1250) — compile-verified
//
#include <hip/hip_runtime.h>
#include <math.h>

#define NUM_SEQS     16
#define MAX_BLOCKS   256
#define N_Q_HEADS    32
#define N_KV_HEADS   8
#define HEAD_SIZE    128
#define VLLM_BS      16
#define SPARSE_BS    64
#define LOCAL_BLOCKS 16
#define VERT_STRIDE  8
#define GQA_R        4        // N_Q_HEADS / N_KV_HEADS
#define NSPLIT       8        // flash-decode length splits per (seq, kv-head)
#define TILE_ELEMS   2048u    // one (phys,kv-head) K or V tile: 128 dims x 16 tokens, f32
#define VROW         17       // padded V row stride in LDS words (bank-conflict-free)

typedef float v2f  __attribute__((ext_vector_type(2)));
typedef float v8f  __attribute__((ext_vector_type(8)));
typedef unsigned int u32x4 __attribute__((ext_vector_type(4)));
typedef int   i32x4 __attribute__((ext_vector_type(4)));
typedef int   i32x8 __attribute__((ext_vector_type(8)));

// horizontal reductions across the 16-lane half-wave that owns one tile row
__device__ __forceinline__ float hmax16(float x) {
  x = fmaxf(x, __shfl_xor(x, 1, 32));
  x = fmaxf(x, __shfl_xor(x, 2, 32));
  x = fmaxf(x, __shfl_xor(x, 4, 32));
  x = fmaxf(x, __shfl_xor(x, 8, 32));
  return x;
}
__device__ __forceinline__ float hsum16(float x) {
  x += __shfl_xor(x, 1, 32);
  x += __shfl_xor(x, 2, 32);
  x += __shfl_xor(x, 4, 32);
  x += __shfl_xor(x, 8, 32);
  return x;
}

// Low 32 bits of a flat shared-aperture address are the LDS byte offset.
__device__ __forceinline__ unsigned lds_offset_of(const void* p) {
  return (unsigned)(uintptr_t)p;
}

// Issue one TDM descriptor: contiguous 1-D tile of TILE_ELEMS f32 from global
// memory into LDS. pad_v adds 1 DWORD of LDS padding every 16 DWORDs so the
// V tile lands with a 17-word row stride (conflict-free ds reads).
__device__ __forceinline__ void tdm_load_tile(const float* gsrc, unsigned lds_off,
                                              bool pad_v) {
  const unsigned long long ga = (unsigned long long)(uintptr_t)gsrc;
  u32x4 g0;
  g0.x = 1u;                                            // count=1 (valid user D#)
  g0.y = lds_off;                                       // lds_addr (bytes)
  g0.z = (unsigned)ga;                                  // global_addr[31:0]
  g0.w = (unsigned)((ga >> 32) & 0x01FFFFFFu) | (2u << 30); // [56:32] | type=2
  i32x8 g1;
  unsigned w0 = 2u << 16;                               // data_size = 4 bytes
  if (pad_v) w0 |= (1u << 20) | (3u << 22);             // pad_enable, 16 DW interval, +1 DW
  g1[0] = (int)w0;                                      // workgroup_mask=0 (no cluster)
  g1[1] = (int)(TILE_ELEMS << 16);                      // tensor_dim0[15:0] @ [31:16]
  g1[2] = (int)((TILE_ELEMS >> 16) | (1u << 16));       // dim0[31:16] | tensor_dim1=1
  g1[3] = (int)(TILE_ELEMS << 16);                      // dim1 hi=0 | tile_dim0=2048
  g1[4] = 1;                                            // tile_dim1=1, tile_dim2=0
  g1[5] = (int)TILE_ELEMS;                              // tensor_dim0_stride[31:0]
  g1[6] = (int)(TILE_ELEMS << 16);                      // stride0 hi=0 | stride1[15:0]
  g1[7] = 0;                                            // stride1[47:16]
  const i32x4 gz4 = {0, 0, 0, 0};
  const i32x8 gz8 = {0, 0, 0, 0, 0, 0, 0, 0};
  // 6-arg (clang-23 / therock) form: (g0, g1, g2, g3, g4, cpol)
  __builtin_amdgcn_tensor_load_to_lds(g0, g1, gz4, gz4, gz8, 0);
}

// First tile index >= T in [.., t1) that any of the group's 4 heads needs.
__device__ __forceinline__ int next_needed(int T, int t1, int qb, int h) {
  for (; T < t1; ++T) {
    const int kb = T >> 2;
    bool need = (qb - kb) < LOCAL_BLOCKS;
    #pragma unroll
    for (int r = 0; r < GQA_R; ++r)
      need = need || (((kb + h * GQA_R + r + 1) & (VERT_STRIDE - 1)) == 0);
    if (need) return T;
  }
  return t1;
}

// One wave per (split, kv_head, seq). Flash attention over the split's range
// of 16-token vllm blocks; 4 GQA q-heads ride as rows 0..3 of the 16x16 WMMA
// tiles. K/V tiles are streamed into double-buffered LDS by the TDM, one tile
// ahead of compute; sparse-block skip happens BEFORE the DMA is even issued.
__global__ __launch_bounds__(32)
void sparse_paged_attn_partial(const float* __restrict__ q,
                               const float* __restrict__ k_cache,
                               const float* __restrict__ v_cache,
                               const int*   __restrict__ block_tables,
                               const int*   __restrict__ ctx_lens,
                               float* __restrict__ ws_acc,   // [S][Hkv][NSPLIT][4][128]
                               float* __restrict__ ws_ml)    // [S][Hkv][NSPLIT][4][2]
{
  const int split = blockIdx.x;
  const int h     = blockIdx.y;   // kv head
  const int s     = blockIdx.z;   // sequence
  const int lane  = threadIdx.x;
  const int half  = lane >> 4;    // 0: K-rows {0,1}, 1: K-rows {2,3} of x4 fragment
  const int lm    = lane & 15;

  __shared__ float Qs[16][HEAD_SIZE];       // A-tile rows (rows 4..15 stay zero)
  __shared__ float Ps[16][16];              // P A-tile for PV (rows 4..15 stay zero)
  __shared__ float Ks[2][TILE_ELEMS];       // K tile: [d][token], row stride 16
  __shared__ float Vs[2][HEAD_SIZE * VROW]; // V tile padded: row stride 17

  for (int i = lane; i < 16 * HEAD_SIZE; i += 32) (&Qs[0][0])[i] = 0.0f;
  for (int i = lane; i < 16 * 16;        i += 32) (&Ps[0][0])[i] = 0.0f;
  __syncthreads();
  for (int i = lane; i < GQA_R * HEAD_SIZE; i += 32) {
    const int r = i >> 7, d = i & (HEAD_SIZE - 1);
    Qs[r][d] = q[((size_t)s * N_Q_HEADS + h * GQA_R + r) * HEAD_SIZE + d];
  }
  __syncthreads();

  const int ctx  = ctx_lens[s];
  const int qpos = ctx - 1;
  const int qb   = qpos >> 6;                 // query sparse-block index
  const int nb   = (ctx + VLLM_BS - 1) >> 4;  // #vllm blocks with valid keys
  const int chunk = (nb + NSPLIT - 1) / NSPLIT;
  const int t0 = split * chunk;
  const int t1 = min(nb, t0 + chunk);

  float m0 = -INFINITY, m1 = -INFINITY, m2 = -INFINITY, m3 = -INFINITY;
  float l0 = 0.f, l1 = 0.f, l2 = 0.f, l3 = 0.f;

  v8f zero8;
  #pragma unroll
  for (int i = 0; i < 8; ++i) zero8[i] = 0.0f;
  v8f acc[8];           // out rows 0..3 x 128 dims, C-layout per 16-dim chunk
  #pragma unroll
  for (int dc = 0; dc < 8; ++dc) acc[dc] = zero8;

  const float scale = 0.08838834764831845f;   // 1/sqrt(128)

  // ---- TDM double-buffer pipeline over needed tiles ------------------------
  int Tc  = next_needed(t0, t1, qb, h);
  int buf = 0;
  if (Tc < t1) {
    const int phys = block_tables[s * MAX_BLOCKS + Tc];
    const size_t off = ((size_t)phys * N_KV_HEADS + h) * (HEAD_SIZE * VLLM_BS);
    tdm_load_tile(k_cache + off, lds_offset_of(&Ks[0][0]), false);
    tdm_load_tile(v_cache + off, lds_offset_of(&Vs[0][0]), true);
  }

  while (Tc < t1) {
    const int Tn = next_needed(Tc + 1, t1, qb, h);
    if (Tn < t1) {
      const int physn = block_tables[s * MAX_BLOCKS + Tn];
      const size_t offn = ((size_t)physn * N_KV_HEADS + h) * (HEAD_SIZE * VLLM_BS);
      tdm_load_tile(k_cache + offn, lds_offset_of(&Ks[buf ^ 1][0]), false);
      tdm_load_tile(v_cache + offn, lds_offset_of(&Vs[buf ^ 1][0]), true);
      __builtin_amdgcn_s_wait_tensorcnt(2);   // current buf's 2 DMAs done (in-order)
    } else {
      __builtin_amdgcn_s_wait_tensorcnt(0);
    }
    __syncthreads();                          // compiler ordering for LDS reads

    const float* kb_ = &Ks[buf][0];
    const float* vb_ = &Vs[buf][0];
    const int    kb  = Tc >> 2;               // 64-token sparse block index

    // ---------------- S = Q x K  (16 head-rows x 16 keys) -------------------
    v8f c = zero8;
    #pragma unroll
    for (int kd = 0; kd < HEAD_SIZE / 4; ++kd) {
      const int d0 = kd * 4 + half * 2;
      v2f a, b;
      a.x = Qs[lm][d0];
      a.y = Qs[lm][d0 + 1];
      b.x = kb_[d0 * VLLM_BS + lm];           // K[d0 ][token=lm] from LDS
      b.y = kb_[(d0 + 1) * VLLM_BS + lm];     // K[d0+1][token=lm]
      c = __builtin_amdgcn_wmma_f32_16x16x4_f32(false, a, false, b,
                                                (short)0, c, false, false);
    }

    // ---------------- mask + online softmax for rows 0..3 -------------------
    const int  j      = Tc * VLLM_BS + lm;    // this lane's key position
    const bool causal = (j <= qpos);
    const bool local  = (qb - kb) < LOCAL_BLOCKS;
    float p0, p1, p2, p3, cr0, cr1, cr2, cr3;

#define FLASH_ROW(R, mR, lR, pR, crR)                                          \
    {                                                                          \
      const bool vert = (((kb + h * GQA_R + R + 1) & (VERT_STRIDE - 1)) == 0); \
      float sc = (causal && (local || vert)) ? c[R] * scale : -INFINITY;       \
      const float rm = hmax16(sc);                                             \
      const float mn = fmaxf(mR, rm);                                          \
      crR = (mn == -INFINITY) ? 1.0f : __expf(mR - mn);                        \
      pR  = (sc == -INFINITY) ? 0.0f : __expf(sc - mn);                        \
      lR  = lR * crR + hsum16(pR);                                             \
      mR  = mn;                                                                \
    }
    FLASH_ROW(0, m0, l0, p0, cr0)
    FLASH_ROW(1, m1, l1, p1, cr1)
    FLASH_ROW(2, m2, l2, p2, cr2)
    FLASH_ROW(3, m3, l3, p3, cr3)
#undef FLASH_ROW

    if (lane < 16) {          // C-layout -> A-layout transpose through LDS
      Ps[0][lm] = p0;
      Ps[1][lm] = p1;
      Ps[2][lm] = p2;
      Ps[3][lm] = p3;
    }
    #pragma unroll
    for (int dc = 0; dc < 8; ++dc) {          // flash rescale (rows 4..7 unused)
      acc[dc][0] *= cr0;
      acc[dc][1] *= cr1;
      acc[dc][2] *= cr2;
      acc[dc][3] *= cr3;
    }
    __syncthreads();

    // ---------------- O += P x V  (rows x 128 dims, 8 chunks of 16) ---------
    #pragma unroll
    for (int dc = 0; dc < 8; ++dc) {
      const int dim = dc * 16 + lm;
      #pragma unroll
      for (int t = 0; t < 4; ++t) {
        const int k0 = t * 4 + half * 2;      // key index inside tile
        v2f a, b;
        a.x = Ps[lm][k0];
        a.y = Ps[lm][k0 + 1];
        b.x = vb_[dim * VROW + k0];           // padded row stride: conflict-free
        b.y = vb_[dim * VROW + k0 + 1];
        acc[dc] = __builtin_amdgcn_wmma_f32_16x16x4_f32(false, a, false, b,
                                                        (short)0, acc[dc],
                                                        false, false);
      }
    }
    __syncthreads();          // protect Ps before next tile rewrites it

    buf ^= 1;
    Tc = Tn;
  }

  // ---------------- write split partial ------------------------------------
  const size_t base = ((size_t)s * N_KV_HEADS + h) * NSPLIT + split;
  if (lane < 16) {
    #pragma unroll
    for (int dc = 0; dc < 8; ++dc) {
      #pragma unroll
      for (int r = 0; r < GQA_R; ++r)
        ws_acc[(base * GQA_R + r) * HEAD_SIZE + dc * 16 + lm] = acc[dc][r];
    }
  }
  if (lane == 0) {
    ws_ml[(base * GQA_R + 0) * 2] = m0; ws_ml[(base * GQA_R + 0) * 2 + 1] = l0;
    ws_ml[(base * GQA_R + 1) * 2] = m1; ws_ml[(base * GQA_R + 1) * 2 + 1] = l1;
    ws_ml[(base * GQA_R + 2) * 2] = m2; ws_ml[(base * GQA_R + 2) * 2 + 1] = l2;
    ws_ml[(base * GQA_R + 3) * 2] = m3; ws_ml[(base * GQA_R + 3) * 2 + 1] = l3;
  }
}

// log-sum-exp merge of the NSPLIT partials; one block per (seq, q-head),
// one thread per head dim.
__global__ __launch_bounds__(HEAD_SIZE)
void sparse_paged_attn_combine(const float* __restrict__ ws_acc,
                               const float* __restrict__ ws_ml,
                               float* __restrict__ out)
{
  const int hq = blockIdx.x & (N_Q_HEADS - 1);
  const int s  = blockIdx.x >> 5;
  const int g  = hq >> 2;          // kv head
  const int r  = hq & 3;           // row within group
  const int d  = threadIdx.x;

  const size_t base = ((size_t)s * N_KV_HEADS + g) * NSPLIT;
  float mmax = -INFINITY;
  #pragma unroll
  for (int i = 0; i < NSPLIT; ++i)
    mmax = fmaxf(mmax, ws_ml[((base + i) * GQA_R + r) * 2]);

  float L = 0.0f, O = 0.0f;
  #pragma unroll
  for (int i = 0; i < NSPLIT; ++i) {
    const float mi = ws_ml[((base + i) * GQA_R + r) * 2];
    const float li = ws_ml[((base + i) * GQA_R + r) * 2 + 1];
    const float w  = (li > 0.0f) ? __expf(mi - mmax) : 0.0f;
    L += w * li;
    O += w * ws_acc[((base + i) * GQA_R + r) * HEAD_SIZE + d];
  }
  out[((size_t)s * N_Q_HEADS + hq) * HEAD_SIZE + d] = (L > 0.0f) ? (O / L) : 0.0f;
}

extern "C" void kernel_launch(void* const* d_in, const int* in_sizes, int n_in,
                              void* d_out, int out_size, void* d_ws, size_t ws_size,
                              hipStream_t stream) {
  const float* q        = (const float*)d_in[0];
  const float* k_cache  = (const float*)d_in[1];
  const float* v_cache  = (const float*)d_in[2];
  const int*   btables  = (const int*)d_in[3];
  const int*   ctx_lens = (const int*)d_in[4];

  float* ws_acc = (float*)d_ws;                                   // 2 MB
  float* ws_ml  = ws_acc +
      (size_t)NUM_SEQS * N_KV_HEADS * NSPLIT * GQA_R * HEAD_SIZE; // +32 KB

  dim3 grid1(NSPLIT, N_KV_HEADS, NUM_SEQS);
  sparse_paged_attn_partial<<<grid1, 32, 0, stream>>>(
      q, k_cache, v_cache, btables, ctx_lens, ws_acc, ws_ml);

  sparse_paged_attn_combine<<<NUM_SEQS * N_Q_HEADS, HEAD_SIZE, 0, stream>>>(
      ws_acc, ws_ml, (float*)d_out);
}